// Temporal_MoE_C_25993142075451
// MI455X (gfx1250) — compile-verified
//
#include <hip/hip_runtime.h>
#include <hip/hip_bf16.h>

typedef __attribute__((ext_vector_type(2))) float v2f;
typedef __attribute__((ext_vector_type(4))) float v4f;
typedef __attribute__((ext_vector_type(8))) float v8f;

#define Hn 2048
#define En 8
#define Tn 2048
#define NTOK (32 * 2048)

// ---------------------------------------------------------------------------
// Kernel 0: G1[r][e] = sum_h time_emb[r,h] * w_gate[h,e]   (33 x 8 table)
// Collapses the entire interval-embedding half of the gate GEMM.
// ---------------------------------------------------------------------------
__global__ void tmoe_g1_kernel(const float* __restrict__ time_emb,
                               const float* __restrict__ w_gate,
                               float* __restrict__ G1) {
  __shared__ float red[256];
  const int r = blockIdx.x;      // 0..32
  const int t = threadIdx.x;     // 0..255
  float acc[En];
#pragma unroll
  for (int e = 0; e < En; ++e) acc[e] = 0.0f;
  for (int h = t; h < Hn; h += 256) {
    float te = time_emb[r * Hn + h];
#pragma unroll
    for (int e = 0; e < En; ++e) acc[e] += te * w_gate[h * En + e];
  }
  for (int e = 0; e < En; ++e) {
    red[t] = acc[e];
    __syncthreads();
    for (int s = 128; s > 0; s >>= 1) {
      if (t < s) red[t] += red[t + s];
      __syncthreads();
    }
    if (t == 0) G1[r * En + e] = red[0];
    __syncthreads();
  }
}

// ---------------------------------------------------------------------------
// Kernel 1: gating. One wave handles a tile of 16 tokens.
// logits_abs(16x16) = A(16x4 cosines) x B(4x16 gate cols) accumulated over H
// via V_WMMA_F32_16X16X4_F32 (512 WMMAs / wave). Then + G1[idx] + bias,
// softmax over the 8 valid experts, route written to workspace.
// Pad lanes (N>=8) read a clamped in-bounds column and multiply by 0 so the
// loop body stays branch-free (EXEC all-ones for WMMA, no saveexec/xcnt).
// ---------------------------------------------------------------------------
__global__ void tmoe_gate_kernel(const float* __restrict__ ts,
                                 const float* __restrict__ w_ab,
                                 const float* __restrict__ b_ab,
                                 const float* __restrict__ w_gate,
                                 const float* __restrict__ b_gate,
                                 const float* __restrict__ G1,
                                 float* __restrict__ route) {
  __shared__ float lgt[8][256];               // 16x16 logit tile per wave
  const int lane = threadIdx.x & 31;
  const int wave = threadIdx.x >> 5;
  const int tile = blockIdx.x * 8 + wave;
  const int g0   = tile * 16;                 // first token of tile
  const int m    = lane & 15;                 // token row within tile / B col
  const int kb   = (lane >> 4) * 2;           // K sub-base: lanes 0-15 -> K0,1 ; 16-31 -> K2,3
  const int tok  = g0 + m;

  const float tsm = ts[tok];
  const float* wg2 = w_gate + (size_t)Hn * En;   // lower half: rows H..2H-1
  const int   nc    = m & (En - 1);              // clamped (in-bounds) column
  const float bmask = (m < En) ? 1.0f : 0.0f;    // zero out pad columns N>=8

  // inv_freq[h] = 10000^(-h/H) = 2^(c*h), maintained incrementally (step of 4)
  const float c  = -13.287712379549449f / (float)Hn;  // -log2(10000)/H
  float if0 = __builtin_amdgcn_exp2f(c * (float)kb);
  float if1 = __builtin_amdgcn_exp2f(c * (float)(kb + 1));
  const float r4 = __builtin_amdgcn_exp2f(c * 4.0f);

  v8f acc = {};
#pragma unroll 4
  for (int h0 = 0; h0 < Hn; h0 += 4) {
    const int ha = h0 + kb;
    const int hb = ha + 1;
    v2f a, b;
    a.x = __cosf((tsm * w_ab[ha] + b_ab[ha]) * if0);
    a.y = __cosf((tsm * w_ab[hb] + b_ab[hb]) * if1);
    b.x = wg2[(size_t)ha * En + nc] * bmask;   // unconditional load + mask
    b.y = wg2[(size_t)hb * En + nc] * bmask;
    acc = __builtin_amdgcn_wmma_f32_16x16x4_f32(false, a, false, b,
                                                (short)0, acc, false, false);
    if0 *= r4;
    if1 *= r4;
  }

  // C/D layout: VGPR r holds M = r + 8*(lane>>4), N = lane&15
  {
    const int mhalf = (lane >> 4) * 8;
    const int nn = lane & 15;
#pragma unroll
    for (int r = 0; r < 8; ++r)
      lgt[wave][(mhalf + r) * 16 + nn] = acc[r];
  }
  __syncthreads();

  if (lane < 16) {
    // interval index for this token
    int idx = 0;
    if ((tok & (Tn - 1)) != 0) {
      float dt = tsm - ts[tok - 1];
      idx = (int)floorf(__log2f(dt + 1.0f));
      if (idx < 0) idx = 0;
      if (idx > 32) idx = 32;
    }
    float l[En];
    float mx = -3.0e38f;
#pragma unroll
    for (int e = 0; e < En; ++e) {
      l[e] = lgt[wave][m * 16 + e] + G1[idx * En + e] + b_gate[e];
      mx = fmaxf(mx, l[e]);
    }
    float s = 0.0f;
#pragma unroll
    for (int e = 0; e < En; ++e) {
      l[e] = __expf(l[e] - mx);
      s += l[e];
    }
    const float inv = 1.0f / s;
    float* rp = route + (size_t)tok * En;
#pragma unroll
    for (int e = 0; e < En; ++e) rp[e] = l[e] * inv;
  }
}

// ---------------------------------------------------------------------------
// Kernel 2 (bandwidth kernel): out = vector * (route @ experts).
// Each thread owns a fixed 8-wide H slice; its 64 expert values live in
// registers and are reused across all tokens (grid-stride). Streaming
// traffic is exactly vector-in + out-out, marked non-temporal.
// ---------------------------------------------------------------------------
__global__ void tmoe_apply_kernel(const float* __restrict__ vec,
                                  const float* __restrict__ route,
                                  const float* __restrict__ experts,
                                  float* __restrict__ out) {
  const int t  = threadIdx.x;     // 0..255
  const int h0 = t * 8;           // 256 threads * 8 = H

  v4f ex0[En], ex1[En];
#pragma unroll
  for (int e = 0; e < En; ++e) {
    const float* p = experts + (size_t)e * Hn + h0;
    ex0[e] = *(const v4f*)p;
    ex1[e] = *(const v4f*)(p + 4);
  }

  for (int tok = blockIdx.x; tok < NTOK; tok += gridDim.x) {
    const float* rp = route + (size_t)tok * En;
    v4f s0 = {}, s1 = {};
#pragma unroll
    for (int e = 0; e < En; ++e) {
      const float r = rp[e];           // block-uniform -> scalar loads
      s0 += r * ex0[e];
      s1 += r * ex1[e];
    }
    const float* vp = vec + (size_t)tok * Hn + h0;
    v4f v0 = __builtin_nontemporal_load((const v4f*)vp);
    v4f v1 = __builtin_nontemporal_load((const v4f*)vp + 1);
    float* op = out + (size_t)tok * Hn + h0;
    __builtin_nontemporal_store(v0 * s0, (v4f*)op);
    __builtin_nontemporal_store(v1 * s1, (v4f*)op + 1);
  }
}

// ---------------------------------------------------------------------------
extern "C" void kernel_launch(void* const* d_in, const int* in_sizes, int n_in,
                              void* d_out, int out_size, void* d_ws, size_t ws_size,
                              hipStream_t stream) {
  const float* vector    = (const float*)d_in[0];
  const float* timestamp = (const float*)d_in[1];
  const float* w_ab      = (const float*)d_in[2];
  const float* b_ab      = (const float*)d_in[3];
  const float* w_gate    = (const float*)d_in[4];
  const float* b_gate    = (const float*)d_in[5];
  const float* time_emb  = (const float*)d_in[6];
  const float* experts   = (const float*)d_in[7];
  float* out = (float*)d_out;

  float* G1    = (float*)d_ws;                      // 33*8 floats
  float* route = (float*)((char*)d_ws + 4096);      // NTOK*8 floats = 2 MiB

  // Phase 0: 33x8 interval-gate table (tiny)
  tmoe_g1_kernel<<<33, 256, 0, stream>>>(time_emb, w_gate, G1);
  // Phase 1: WMMA gating + softmax -> route   (4096 tiles, 8 waves/block)
  tmoe_gate_kernel<<<NTOK / 16 / 8, 256, 0, stream>>>(
      timestamp, w_ab, b_ab, w_gate, b_gate, G1, route);
  // Phase 2: memory-bound apply (~1 GB streamed, NT hints)
  tmoe_apply_kernel<<<2048, 256, 0, stream>>>(vector, route, experts, out);
}